// RoiPooling_39814346834532
// MI455X (gfx1250) — compile-verified
//
#include <hip/hip_runtime.h>
#include <hip/hip_bf16.h>

#define POOLW   7
#define NROIS   128
#define IMH     28
#define IMW     28
#define IMC     512
#define NPIX    (2048 * POOLW * POOLW)   // 100352 (roi, py, px) pixels
#define WAVES_PER_BLOCK 8

typedef float vf4 __attribute__((ext_vector_type(4)));   // native vector: OK for nontemporal builtins

// One wave32 per output pixel (roi, py, px): box math is wave-uniform,
// lanes stream the 512-channel vector as coalesced 128-bit accesses.
__global__ __launch_bounds__(256) void roi_bilinear_kernel(
    const float* __restrict__ img,    // img[0]: (28, 28, 512)
    const float* __restrict__ rois,   // flat (2048 * 5)
    float* __restrict__ out)          // flat (2048, 7, 7, 512)
{
    const int lane  = threadIdx.x & 31;
    const int wave  = threadIdx.x >> 5;
    const int pixel = blockIdx.x * WAVES_PER_BLOCK + wave;   // < 100352

    const int n   = pixel / (POOLW * POOLW);
    const int rem = pixel - n * (POOLW * POOLW);
    const int py  = rem / POOLW;
    const int px  = rem - py * POOLW;

    // Reference: r = rois.reshape(-1,5)[:,3:]; x1=col3, y1=col4
    const float bx1 = rois[n * 5 + 3];
    const float by1 = rois[n * 5 + 4];
    // Mimic reference fp32 rounding: x2 = x1 + 4/28; step = (x2-x1)*27/6
    const float bx2 = bx1 + 4.0f / 28.0f;
    const float by2 = by1 + 4.0f / 28.0f;
    const float sx  = (bx2 - bx1) * 27.0f / 6.0f;
    const float sy  = (by2 - by1) * 27.0f / 6.0f;
    const float xs  = bx1 * 27.0f + (float)px * sx;
    const float ys  = by1 * 27.0f + (float)py * sy;

    float* dst = out + (size_t)pixel * IMC;

    // tf crop_and_resize mask: whole sample is zeroed when outside [0, H-1]
    const bool valid = (ys >= 0.0f) && (ys <= 27.0f) &&
                       (xs >= 0.0f) && (xs <= 27.0f);
    if (!valid) {  // wave-uniform branch: skip all gathers
        const vf4 z = (vf4)(0.0f);
#pragma unroll
        for (int j = 0; j < 4; ++j) {
            const int coff = (j * 32 + lane) * 4;
            __builtin_nontemporal_store(z, (vf4*)(dst + coff));
        }
        return;
    }

    const float y0f = floorf(ys);
    const float x0f = floorf(xs);
    const float ly  = ys - y0f;
    const float lx  = xs - x0f;
    const float oly = 1.0f - ly;

    const int y0 = (int)fminf(fmaxf(y0f,        0.0f), 27.0f);
    const int y1 = (int)fminf(fmaxf(y0f + 1.0f, 0.0f), 27.0f);
    const int x0 = (int)fminf(fmaxf(x0f,        0.0f), 27.0f);
    const int x1 = (int)fminf(fmaxf(x0f + 1.0f, 0.0f), 27.0f);

    const float* tlB = img + ((size_t)(y0 * IMW + x0)) * IMC;
    const float* trB = img + ((size_t)(y0 * IMW + x1)) * IMC;
    const float* blB = img + ((size_t)(y1 * IMW + x0)) * IMC;
    const float* brB = img + ((size_t)(y1 * IMW + x1)) * IMC;

    // gfx1250 prefetch: lane-spread at 64B covers each 2KB corner row
    __builtin_prefetch(tlB + lane * 16, 0, 3);
    __builtin_prefetch(trB + lane * 16, 0, 3);
    __builtin_prefetch(blB + lane * 16, 0, 3);
    __builtin_prefetch(brB + lane * 16, 0, 3);

#pragma unroll
    for (int j = 0; j < 4; ++j) {
        const int coff = (j * 32 + lane) * 4;   // contiguous 512B per wave op
        const vf4 tl = *(const vf4*)(tlB + coff);
        const vf4 tr = *(const vf4*)(trB + coff);
        const vf4 bl = *(const vf4*)(blB + coff);
        const vf4 br = *(const vf4*)(brB + coff);
        const vf4 top = tl + (tr - tl) * lx;
        const vf4 bot = bl + (br - bl) * lx;
        const vf4 v   = top * oly + bot * ly;
        // streaming output (205MB, write-once): non-temporal B128 store
        __builtin_nontemporal_store(v, (vf4*)(dst + coff));
    }
}

extern "C" void kernel_launch(void* const* d_in, const int* in_sizes, int n_in,
                              void* d_out, int out_size, void* d_ws, size_t ws_size,
                              hipStream_t stream) {
    (void)in_sizes; (void)n_in; (void)out_size; (void)d_ws; (void)ws_size;
    const float* img  = (const float*)d_in[0];   // (16,28,28,512) -> only img[0] used
    const float* rois = (const float*)d_in[1];   // (16, 640) -> flat (2048,5)
    float* out = (float*)d_out;                  // 51,380,224 floats

    dim3 block(256);
    dim3 grid(NPIX / WAVES_PER_BLOCK);           // 12544, exact
    hipLaunchKernelGGL(roi_bilinear_kernel, grid, block, 0, stream, img, rois, out);
}